// deepwideNet_76390288327751
// MI455X (gfx1250) — compile-verified
//
#include <hip/hip_runtime.h>

#define NEG_SLOPE 0.01f

typedef float v2f __attribute__((ext_vector_type(2)));
typedef float v8f __attribute__((ext_vector_type(8)));

// ---------------- utility kernels ----------------

__global__ void zero4_kernel(float4* __restrict__ p, int n4) {
    int i = blockIdx.x * blockDim.x + threadIdx.x;
    if (i < n4) p[i] = make_float4(0.f, 0.f, 0.f, 0.f);
}

__global__ void zero_kernel(float* __restrict__ p, int n) {
    int i = blockIdx.x * blockDim.x + threadIdx.x;
    if (i < n) p[i] = 0.0f;
}

__global__ void degree_kernel(const int* __restrict__ src, const int* __restrict__ dst,
                              float* __restrict__ dout, float* __restrict__ din, int e) {
    int i = blockIdx.x * blockDim.x + threadIdx.x;
    if (i < e) {
        atomicAdd(&dout[src[i]], 1.0f);
        atomicAdd(&din[dst[i]], 1.0f);
    }
}

__global__ void invsqrt_kernel(float* __restrict__ d, int n) {
    int i = blockIdx.x * blockDim.x + threadIdx.x;
    if (i < n) d[i] = rsqrtf(fmaxf(d[i], 1.0f));
}

// ---------------- WMMA GEMM:  H = (X * dout_is[:,None]) @ W ----------------
// One wave computes one 16x16 output tile with V_WMMA_F32_16X16X4_F32.
// A 16x4 f32 fragment: V0 = K(2*half+0), V1 = K(2*half+1), row = lane&15.
// B 4x16 f32 fragment: mirrored (same K per VGPR, col = lane&15).
// D 16x16 f32: VGPR r -> row (8*half + r), col = lane&15.
template<int FI, int FO>
__global__ __launch_bounds__(256) void gemm_wmma_kernel(
        const float* __restrict__ X, const float* __restrict__ dout_is,
        const float* __restrict__ W, float* __restrict__ H, int nTilesM) {
    constexpr int TILES_N = (FO + 15) / 16;
    int wave = (blockIdx.x * blockDim.x + threadIdx.x) >> 5;
    if (wave >= nTilesM * TILES_N) return;   // wave-uniform exit; EXEC all-1s at WMMA
    int lane = threadIdx.x & 31;
    int tm = wave / TILES_N;
    int tn = wave - tm * TILES_N;
    int half = lane >> 4;
    int l15  = lane & 15;
    int row  = tm * 16 + l15;
    int col  = tn * 16 + l15;
    float s  = dout_is[row];

    v8f acc = {};
#pragma unroll
    for (int kb = 0; kb < FI; kb += 4) {
        int k = kb + 2 * half;
        v2f a, b;
        a.x = X[row * FI + k]     * s;
        a.y = X[row * FI + k + 1] * s;
        if (col < FO) {
            b.x = W[k * FO + col];
            b.y = W[(k + 1) * FO + col];
        } else {
            b.x = 0.0f; b.y = 0.0f;
        }
        acc = __builtin_amdgcn_wmma_f32_16x16x4_f32(
            /*neg_a=*/false, a, /*neg_b=*/false, b,
            /*c_mod=*/(short)0, acc, /*reuse_a=*/false, /*reuse_b=*/false);
    }
    if (col < FO) {
#pragma unroll
        for (int r = 0; r < 8; ++r)
            H[(tm * 16 + half * 8 + r) * FO + col] = acc[r];
    }
}

// Scalar fallback for row-tail (unused when N % 16 == 0, e.g. N=100000).
__global__ void gemm_tail_kernel(const float* __restrict__ X, const float* __restrict__ dout_is,
                                 const float* __restrict__ W, float* __restrict__ H,
                                 int rowStart, int nRows, int fi, int fo) {
    int idx = blockIdx.x * blockDim.x + threadIdx.x;
    if (idx >= nRows * fo) return;
    int r = rowStart + idx / fo;
    int c = idx % fo;
    float s = dout_is[r];
    float acc = 0.0f;
    for (int k = 0; k < fi; ++k) acc += X[r * fi + k] * s * W[k * fo + c];
    H[r * fo + c] = acc;
}

// ---------------- edge scatter-add:  agg[dst] += h[src] ----------------
// FO=128: one wave per edge, each lane owns 4 contiguous channels:
// one global_load_b128 + 4 global_atomic_add_f32 per lane.
__global__ __launch_bounds__(256) void scatter128_kernel(
        const float4* __restrict__ H4, const int* __restrict__ src,
        const int* __restrict__ dst, float* __restrict__ agg, int e) {
    long long idx = (long long)blockIdx.x * blockDim.x + threadIdx.x;
    int edge = (int)(idx >> 5);
    if (edge >= e) return;
    int lane = (int)(idx & 31);
    int s = src[edge];
    int d = dst[edge];
    float4 v = H4[s * 32 + lane];            // row stride = 128 floats = 32 float4
    float* a = agg + d * 128 + lane * 4;
    atomicAdd(a + 0, v.x);
    atomicAdd(a + 1, v.y);
    atomicAdd(a + 2, v.z);
    atomicAdd(a + 3, v.w);
}

// Generic (used for FO=3)
template<int FO>
__global__ __launch_bounds__(256) void scatter_kernel(
        const float* __restrict__ H, const int* __restrict__ src,
        const int* __restrict__ dst, float* __restrict__ agg, int e) {
    long long idx = (long long)blockIdx.x * blockDim.x + threadIdx.x;
    if (idx >= (long long)e * FO) return;
    int edge = (int)(idx / FO);
    int c    = (int)(idx - (long long)edge * FO);
    atomicAdd(&agg[dst[edge] * FO + c], H[src[edge] * FO + c]);
}

// ---------------- finalize: out = lrelu(agg * din_is + b) ----------------
__device__ __forceinline__ float lrelu1(float v, bool relu) {
    return (!relu || v >= 0.0f) ? v : NEG_SLOPE * v;
}

// FO=128 vectorized: one thread per float4 (4 channels).
template<bool RELU>
__global__ void finalize128_kernel(const float4* __restrict__ agg4,
                                   const float* __restrict__ din_is,
                                   const float4* __restrict__ bias4,
                                   float4* __restrict__ out4, int n) {
    int idx = blockIdx.x * blockDim.x + threadIdx.x;
    if (idx >= n * 32) return;
    int i  = idx >> 5;
    int c4 = idx & 31;
    float s  = din_is[i];
    float4 a = agg4[idx];
    float4 b = bias4[c4];
    float4 v;
    v.x = lrelu1(a.x * s + b.x, RELU);
    v.y = lrelu1(a.y * s + b.y, RELU);
    v.z = lrelu1(a.z * s + b.z, RELU);
    v.w = lrelu1(a.w * s + b.w, RELU);
    out4[idx] = v;
}

template<int FO, bool RELU>
__global__ void finalize_kernel(const float* __restrict__ agg, const float* __restrict__ din_is,
                                const float* __restrict__ bias, float* __restrict__ out, int n) {
    int idx = blockIdx.x * blockDim.x + threadIdx.x;
    if (idx >= n * FO) return;
    int i = idx / FO;
    int c = idx - i * FO;
    out[idx] = lrelu1(agg[idx] * din_is[i] + bias[c], RELU);
}

// ---------------- host-side helpers ----------------
static void zero_n(float* p, long long n, hipStream_t stream) {
    long long n4 = n >> 2;
    if (n4 > 0)
        zero4_kernel<<<(int)((n4 + 255) / 256), 256, 0, stream>>>((float4*)p, (int)n4);
    long long tail = n - n4 * 4;
    if (tail > 0)
        zero_kernel<<<1, 256, 0, stream>>>(p + n4 * 4, (int)tail);
}

template<int FI, int FO, bool RELU>
static void run_layer(const float* x, const float* W, const float* b,
                      const float* dout_is, const float* din_is,
                      float* H, float* AGG, float* out,
                      const int* src, const int* dst, int N, int E, hipStream_t stream) {
    int tilesM = N / 16;
    if (tilesM > 0) {
        int nWaves = tilesM * ((FO + 15) / 16);
        int blocks = (nWaves * 32 + 255) / 256;
        gemm_wmma_kernel<FI, FO><<<blocks, 256, 0, stream>>>(x, dout_is, W, H, tilesM);
    }
    int tail = N - tilesM * 16;
    if (tail > 0) {
        int tot = tail * FO;
        gemm_tail_kernel<<<(tot + 255) / 256, 256, 0, stream>>>(x, dout_is, W, H,
                                                                tilesM * 16, tail, FI, FO);
    }
    long long ntot = (long long)N * FO;
    zero_n(AGG, ntot, stream);
    if (FO == 128) {
        long long thr = (long long)E * 32;
        scatter128_kernel<<<(int)((thr + 255) / 256), 256, 0, stream>>>(
            (const float4*)H, src, dst, AGG, E);
        finalize128_kernel<RELU><<<(int)((ntot / 4 + 255) / 256), 256, 0, stream>>>(
            (const float4*)AGG, din_is, (const float4*)b, (float4*)out, N);
    } else {
        long long etot = (long long)E * FO;
        scatter_kernel<FO><<<(int)((etot + 255) / 256), 256, 0, stream>>>(H, src, dst, AGG, E);
        finalize_kernel<FO, RELU><<<(int)((ntot + 255) / 256), 256, 0, stream>>>(
            AGG, din_is, b, out, N);
    }
}

extern "C" void kernel_launch(void* const* d_in, const int* in_sizes, int n_in,
                              void* d_out, int out_size, void* d_ws, size_t ws_size,
                              hipStream_t stream) {
    const float* feat = (const float*)d_in[0];
    const int*   src  = (const int*)d_in[1];
    const int*   dst  = (const int*)d_in[2];
    const float* W1 = (const float*)d_in[3];  const float* b1 = (const float*)d_in[4];
    const float* W2 = (const float*)d_in[5];  const float* b2 = (const float*)d_in[6];
    const float* W3 = (const float*)d_in[7];  const float* b3 = (const float*)d_in[8];
    const float* W4 = (const float*)d_in[9];  const float* b4 = (const float*)d_in[10];
    const float* W5 = (const float*)d_in[11]; const float* b5 = (const float*)d_in[12];

    const int N = in_sizes[0] / 4;
    const int E = in_sizes[1];

    float* ws   = (float*)d_ws;
    float* dout = ws;                              // [N]
    float* din  = ws + N;                          // [N]
    float* P    = ws + 2 * (size_t)N;              // [N,128] activations
    float* Q    = P + (size_t)N * 128;             // [N,128] h = xW
    float* R    = Q + (size_t)N * 128;             // [N,128] agg

    // degrees -> inverse sqrt (dout|din contiguous: one pass for both)
    zero_n(dout, 2LL * N, stream);
    degree_kernel<<<(E + 255) / 256, 256, 0, stream>>>(src, dst, dout, din, E);
    invsqrt_kernel<<<(2 * N + 255) / 256, 256, 0, stream>>>(dout, 2 * N);

    run_layer<4,   128, true >(feat, W1, b1, dout, din, Q, R, P, src, dst, N, E, stream);
    run_layer<128, 128, true >(P,    W2, b2, dout, din, Q, R, P, src, dst, N, E, stream);
    run_layer<128, 128, true >(P,    W3, b3, dout, din, Q, R, P, src, dst, N, E, stream);
    run_layer<128, 128, true >(P,    W4, b4, dout, din, Q, R, P, src, dst, N, E, stream);
    run_layer<128, 3,   false>(P,    W5, b5, dout, din, Q, R, (float*)d_out, src, dst, N, E, stream);
}